// GCN_Loss_1151051235756
// MI455X (gfx1250) — compile-verified
//
#include <hip/hip_runtime.h>
#include <hip/hip_bf16.h>

typedef __attribute__((ext_vector_type(2))) float v2f;
typedef __attribute__((ext_vector_type(8))) float v8f;

#define DIMS 128
#define PART_STRIDE 264   // 128 (sum0) + 128 (sum1) + 1 (ssq) + 1 (cnt0) + pad

// Stage 1: each block of 8 waves grid-strides over rows; one wave streams one
// full 512B row per iteration (lane l loads cols 4l..4l+3 as b128). Per-lane
// masked FMA accumulation (bandwidth-bound: ~3 VALU ops per loaded float).
// Cross-wave reduction of the 8x128 per-class partials is done with
// V_WMMA_F32_16X16X4_F32 using a ones A-matrix (column sums on the matrix pipe).
__global__ void fisher_partial_kernel(const float* __restrict__ X,
                                      const int* __restrict__ lab,
                                      float* __restrict__ part,
                                      int N, int nblocks) {
    __shared__ float lds0[8 * DIMS];
    __shared__ float lds1[8 * DIMS];
    __shared__ float sWave[8];
    __shared__ float cWave[8];

    const int lane = threadIdx.x & 31;
    const int w    = threadIdx.x >> 5;
    const int gw   = blockIdx.x * 8 + w;   // global wave id
    const int TW   = nblocks * 8;          // total waves

    float4 a0 = make_float4(0.f, 0.f, 0.f, 0.f);
    float4 a1 = make_float4(0.f, 0.f, 0.f, 0.f);
    float ssq  = 0.f;
    float cnt0 = 0.f;

    for (int r = gw; r < N; r += TW) {
        const float4 v = reinterpret_cast<const float4*>(X + (size_t)r * DIMS)[lane];
        const int nr = r + TW;
        if (nr < N) {
            // global_prefetch_b8: warm next strided row into the hierarchy
            __builtin_prefetch(reinterpret_cast<const char*>(
                X + (size_t)nr * DIMS + lane * 4), 0, 1);
        }
        const float is0 = (lab[r] < 0) ? 1.0f : 0.0f;   // label == -1
        const float is1 = 1.0f - is0;
        a0.x = fmaf(v.x, is0, a0.x);  a1.x = fmaf(v.x, is1, a1.x);
        a0.y = fmaf(v.y, is0, a0.y);  a1.y = fmaf(v.y, is1, a1.y);
        a0.z = fmaf(v.z, is0, a0.z);  a1.z = fmaf(v.z, is1, a1.z);
        a0.w = fmaf(v.w, is0, a0.w);  a1.w = fmaf(v.w, is1, a1.w);
        ssq = fmaf(v.x, v.x, ssq);
        ssq = fmaf(v.y, v.y, ssq);
        ssq = fmaf(v.z, v.z, ssq);
        ssq = fmaf(v.w, v.w, ssq);
        cnt0 += is0;            // lane-uniform per wave
    }

    // stash wave partials in LDS (columns are partitioned across lanes: no
    // intra-wave reduction needed for the vectors)
    lds0[w * DIMS + lane * 4 + 0] = a0.x;
    lds0[w * DIMS + lane * 4 + 1] = a0.y;
    lds0[w * DIMS + lane * 4 + 2] = a0.z;
    lds0[w * DIMS + lane * 4 + 3] = a0.w;
    lds1[w * DIMS + lane * 4 + 0] = a1.x;
    lds1[w * DIMS + lane * 4 + 1] = a1.y;
    lds1[w * DIMS + lane * 4 + 2] = a1.z;
    lds1[w * DIMS + lane * 4 + 3] = a1.w;

    // ssq needs an intra-wave reduction (wave32)
    for (int off = 16; off > 0; off >>= 1)
        ssq += __shfl_xor(ssq, off, 32);
    if (lane == 0) { sWave[w] = ssq; cWave[w] = cnt0; }

    __syncthreads();

    if (w == 0) {
        // Wave 0: reduce 8 wave-partials x 128 cols per class with f32 WMMA.
        // A = ones (16x4)  =>  D[m,n] = sum_k B[k,n] for every m.
        // B tile (4x16): VGPR0 holds K={0,2}, VGPR1 holds K={1,3} halves.
        const int half = lane >> 4;        // 0: lanes 0-15, 1: lanes 16-31
        const int n    = lane & 15;
        v2f ones; ones.x = 1.0f; ones.y = 1.0f;
        float* dst = part + (size_t)blockIdx.x * PART_STRIDE;

        for (int t = 0; t < 8; ++t) {      // eight 16-column tiles
            v8f c0 = {};
            v8f c1 = {};
            for (int kb = 0; kb < 8; kb += 4) {   // waves 0-3 then 4-7
                const int kA = kb + (half ? 2 : 0);
                const int kB = kb + (half ? 3 : 1);
                v2f b0, b1;
                b0.x = lds0[kA * DIMS + t * 16 + n];
                b0.y = lds0[kB * DIMS + t * 16 + n];
                b1.x = lds1[kA * DIMS + t * 16 + n];
                b1.y = lds1[kB * DIMS + t * 16 + n];
                c0 = __builtin_amdgcn_wmma_f32_16x16x4_f32(
                        false, ones, false, b0, (short)0, c0, false, false);
                c1 = __builtin_amdgcn_wmma_f32_16x16x4_f32(
                        false, ones, false, b1, (short)0, c1, false, false);
            }
            if (lane < 16) {               // row M=0 carries the column sums
                dst[t * 16 + n]        = c0[0];
                dst[DIMS + t * 16 + n] = c1[0];
            }
        }
        if (lane == 0) {
            float s = 0.f, c = 0.f;
            for (int i = 0; i < 8; ++i) { s += sWave[i]; c += cWave[i]; }
            dst[256] = s;
            dst[257] = c;
        }
    }
}

// Stage 2: one block combines the per-block partials into the scalar loss.
__global__ void fisher_final_kernel(const float* __restrict__ part,
                                    float* __restrict__ out,
                                    int G, int N) {
    __shared__ float s0[DIMS];
    __shared__ float s1[DIMS];
    __shared__ float extras[2];   // [0]=ssq, [1]=n0

    const int t = threadIdx.x;
    if (t < DIMS) {
        float x0 = 0.f, x1 = 0.f;
        for (int b = 0; b < G; ++b) {
            x0 += part[(size_t)b * PART_STRIDE + t];
            x1 += part[(size_t)b * PART_STRIDE + DIMS + t];
        }
        s0[t] = x0;
        s1[t] = x1;
    } else if (t == DIMS) {
        float ss = 0.f, c0 = 0.f;
        for (int b = 0; b < G; ++b) {
            ss += part[(size_t)b * PART_STRIDE + 256];
            c0 += part[(size_t)b * PART_STRIDE + 257];
        }
        extras[0] = ss;
        extras[1] = c0;
    }
    __syncthreads();

    if (t == 0) {
        float p0 = 0.f, p1 = 0.f, pt = 0.f;
        for (int i = 0; i < DIMS; ++i) {
            const float u = s0[i];
            const float v = s1[i];
            p0 = fmaf(u, u, p0);
            p1 = fmaf(v, v, p1);
            const float tot = u + v;
            pt = fmaf(tot, tot, pt);
        }
        const float n0  = extras[1];
        const float n1  = (float)N - n0;
        const float ssq = extras[0];
        const float within  = p0 / n0 + p1 / n1;
        const float Sw = ssq - within;
        const float Sb = within - pt / (float)N;
        out[0] = Sw / Sb;
    }
}

extern "C" void kernel_launch(void* const* d_in, const int* in_sizes, int n_in,
                              void* d_out, int out_size, void* d_ws, size_t ws_size,
                              hipStream_t stream) {
    const float* X   = (const float*)d_in[0];
    const int*   lab = (const int*)d_in[1];
    float*       out = (float*)d_out;
    float*       ws  = (float*)d_ws;

    const int ND = in_sizes[0];
    const int N  = ND / DIMS;

    // One partial slot per block (unconditionally written -> no zero-init, no
    // atomics, deterministic). Cap block count by the available workspace.
    const size_t slot_bytes = PART_STRIDE * sizeof(float);
    int G = (int)(ws_size / slot_bytes);
    if (G > 1024) G = 1024;
    if (G < 1)    G = 1;

    fisher_partial_kernel<<<G, 256, 0, stream>>>(X, lab, ws, N, G);
    fisher_final_kernel<<<1, 256, 0, stream>>>(ws, out, G, N);
}